// ConcatAttention_1580547970729
// MI455X (gfx1250) — compile-verified
//
#include <hip/hip_runtime.h>
#include <math.h>

typedef __attribute__((ext_vector_type(2))) float v2f;
typedef __attribute__((ext_vector_type(8))) float v8f;

// gfx1250 has a hardware transcendental TANH (V_TANH_F32). Probe for the
// clang builtin; fall back to libm tanhf (ocml) if this toolchain lacks it.
#if defined(__has_builtin)
#  if __has_builtin(__builtin_amdgcn_tanhf)
#    define FAST_TANH(x) __builtin_amdgcn_tanhf(x)
#  elif __has_builtin(__builtin_amdgcn_tanh_f32)
#    define FAST_TANH(x) __builtin_amdgcn_tanh_f32(x)
#  else
#    define FAST_TANH(x) tanhf(x)
#  endif
#else
#  define FAST_TANH(x) tanhf(x)
#endif

namespace {
constexpr int kB  = 4;
constexpr int kLD = 256;
constexpr int kLE = 256;
constexpr int kD  = 512;   // D_DEC == D_ENC
constexpr int kH  = 256;
constexpr int kNL = 8;
}

// ---------------------------------------------------------------------------
// Phase 1: Out[M,N] = X[M,K] @ W[K,N], fp32, one 16x16 tile per wave using
// V_WMMA_F32_16X16X4_F32.  ISA §7.12.2 layouts (wave32):
//   A 16x4 : lane L holds A[M = L&15][K = 2*(L>>4) + j] in vgpr j
//   B 4x16 : lane L holds B[K = 2*(L>>4) + j][N = L&15] in vgpr j
//   D 16x16: lane L, vgpr r holds D[M = r + 8*(L>>4)][N = L&15]
// ---------------------------------------------------------------------------
__global__ __launch_bounds__(256) void proj_wmma(const float* __restrict__ X,
                                                 const float* __restrict__ W,
                                                 float* __restrict__ Out,
                                                 int M, int K, int N) {
  const int lane  = threadIdx.x & 31;
  const int wave  = threadIdx.x >> 5;
  const int gwave = blockIdx.x * 8 + wave;      // one 16x16 tile per wave
  const int tiles_n = N >> 4;
  const int mt = gwave / tiles_n;
  const int nt = gwave % tiles_n;
  const int l15 = lane & 15;
  const int hi  = lane >> 4;

  const float* xrow = X + (size_t)(mt * 16 + l15) * K;   // A row for this lane
  const float* wcol = W + nt * 16 + l15;                 // B column for this lane

  v8f acc = {};
#pragma unroll 4
  for (int k0 = 0; k0 < K; k0 += 4) {
    const int kk = k0 + 2 * hi;
    v2f a = *(const v2f*)(xrow + kk);                    // A[M=l15][kk..kk+1]
    v2f bfrag;
    bfrag.x = wcol[(size_t)kk * N];                      // B[kk  ][N=l15]
    bfrag.y = wcol[(size_t)(kk + 1) * N];                // B[kk+1][N=l15]
    acc = __builtin_amdgcn_wmma_f32_16x16x4_f32(false, a, false, bfrag,
                                                (short)0, acc, false, false);
  }

  float* orow = Out + (size_t)(mt * 16 + 8 * hi) * N + nt * 16 + l15;
#pragma unroll
  for (int r = 0; r < 8; ++r) orow[(size_t)r * N] = acc[r];
}

// ---------------------------------------------------------------------------
// Phase 2 (fused): for each (b, e) and a 16-row d tile, compute
//   A[d,h] = tanh(Od[b,d,h] + Oe[b,e,h] + bias[h])      (built in-register)
//   energy[b,e,d,l] = sum_h A[d,h] * v[h,l]             (WMMA, N padded 8->16)
// and store directly in the transposed [B, NL, LD, LE] layout.
// Never materializes the 268 MB intermediate of the reference.
// Block = 256 threads (8 waves): one e, half of the d range (8 tiles).
// ---------------------------------------------------------------------------
__global__ __launch_bounds__(256) void fused_energy(const float* __restrict__ Od,
                                                    const float* __restrict__ Oe,
                                                    const float* __restrict__ bias,
                                                    const float* __restrict__ v,
                                                    float* __restrict__ out) {
  __shared__ __align__(16) float sEB[kH];       // Oe[b,e,:] + bias  (precombined)
  __shared__ __align__(16) float sV[kH][17];    // v padded to N=16, stride 17 (banks)

  const int idx   = blockIdx.x;
  const int dhalf = idx & 1;
  const int e     = (idx >> 1) & (kLE - 1);
  const int b     = idx >> 9;                   // / (2*kLE)

  const int tid = threadIdx.x;                  // 256 threads fill 256 h rows
  sEB[tid] = Oe[(size_t)(b * kLE + e) * kH + tid] + bias[tid];
#pragma unroll
  for (int c = 0; c < 16; ++c)
    sV[tid][c] = (c < kNL) ? v[tid * kNL + c] : 0.0f;
  __syncthreads();

  const int lane = tid & 31;
  const int wave = tid >> 5;
  const int l15  = lane & 15;
  const int hi   = lane >> 4;
  const int d_base = (dhalf * 8 + wave) * 16;   // this wave's 16-row d tile

  const float* odrow = Od + (size_t)(b * kLD + d_base + l15) * kH;

  v8f acc = {};
#pragma unroll 4
  for (int k0 = 0; k0 < kH; k0 += 4) {
    const int kk = k0 + 2 * hi;
    v2f od = *(const v2f*)(odrow + kk);         // Od[d=l15][kk..kk+1] (L2-resident)
    v2f eb = *(const v2f*)(&sEB[kk]);           // broadcast within half-wave
    v2f a;
    a.x = FAST_TANH(od.x + eb.x);               // activation = A-fragment
    a.y = FAST_TANH(od.y + eb.y);
    v2f bfrag;
    bfrag.x = sV[kk][l15];                      // v[kk  ][l=l15]
    bfrag.y = sV[kk + 1][l15];                  // v[kk+1][l=l15]
    acc = __builtin_amdgcn_wmma_f32_16x16x4_f32(false, a, false, bfrag,
                                                (short)0, acc, false, false);
  }

  // D[r][lane] -> energy(d = d_base + r + 8*hi, l = l15); store as out[b,l,d,e].
  if (l15 < kNL) {
    float* obase = out + ((size_t)(b * kNL + l15) * kLD + d_base + 8 * hi) * kLE + e;
#pragma unroll
    for (int r = 0; r < 8; ++r) obase[(size_t)r * kLE] = acc[r];
  }
}

extern "C" void kernel_launch(void* const* d_in, const int* in_sizes, int n_in,
                              void* d_out, int out_size, void* d_ws, size_t ws_size,
                              hipStream_t stream) {
  (void)in_sizes; (void)n_in; (void)out_size; (void)ws_size;

  const float* input_d = (const float*)d_in[0];
  const float* input_e = (const float*)d_in[1];
  const float* W_d     = (const float*)d_in[2];
  const float* W_e     = (const float*)d_in[3];
  const float* bias    = (const float*)d_in[4];
  const float* v       = (const float*)d_in[5];
  float*       out     = (float*)d_out;

  // Workspace: Od (1 MiB) + Oe (1 MiB) = 2 MiB of d_ws.
  float* Od = (float*)d_ws;
  float* Oe = Od + (size_t)kB * kLD * kH;

  dim3 blk(256);
  const int tiles = (kB * kLD / 16) * (kH / 16);   // 1024 tiles = 1024 waves
  proj_wmma<<<tiles / 8, blk, 0, stream>>>(input_d, W_d, Od, kB * kLD, kD, kH);
  proj_wmma<<<tiles / 8, blk, 0, stream>>>(input_e, W_e, Oe, kB * kLE, kD, kH);
  fused_energy<<<kB * kLE * 2, blk, 0, stream>>>(Od, Oe, bias, v, out);
}